// DCTLayer_30236569764352
// MI455X (gfx1250) — compile-verified
//
#include <hip/hip_runtime.h>
#include <hip/hip_bf16.h>

// DCT-II(8) -> exact inverse DCT over 8x8 blocks, block-major output.
// Bandwidth-bound (~201 MB @ 23.3 TB/s => ~8.6 us floor). Matmuls on the fp32
// WMMA pipe (v_wmma_f32_16x16x4_f32, K=16 as 4 chained K=4 steps). Stage-in is
// a double-buffered ASYNCcnt pipeline (GLOBAL_LOAD_ASYNC_TO_LDS_B128 issued
// one group ahead, drained with s_wait_asynccnt 2). All LDS tiles are
// wave-private: no workgroup barriers, only wave-local fences (zero ISA cost,
// correctness from CDNA5's in-order same-wave LDS).

typedef float v2f __attribute__((ext_vector_type(2)));
typedef float v4f __attribute__((ext_vector_type(4)));
typedef float v8f __attribute__((ext_vector_type(8)));
typedef int   v4i __attribute__((vector_size(4 * sizeof(int))));

#if defined(__has_builtin)
#  if __has_builtin(__builtin_amdgcn_global_load_async_to_lds_b128)
#    define HAS_ASYNC_LDS 1
#  endif
#  if __has_builtin(__builtin_amdgcn_s_wait_asynccnt)
#    define HAS_WAIT_ASYNC 1
#  endif
#endif

typedef __attribute__((address_space(1))) v4i g_v4i;  // global AS
typedef __attribute__((address_space(3))) v4i l_v4i;  // LDS AS

// drain everything / keep the 2 newest async ops in flight
__device__ __forceinline__ void wait_async_all() {
#if defined(HAS_ASYNC_LDS)
#  if defined(HAS_WAIT_ASYNC)
    __builtin_amdgcn_s_wait_asynccnt(0);
#  else
    asm volatile("s_wait_asynccnt 0x0" ::: "memory");
#  endif
#endif
}
__device__ __forceinline__ void wait_async_keep2() {
#if defined(HAS_ASYNC_LDS)
#  if defined(HAS_WAIT_ASYNC)
    __builtin_amdgcn_s_wait_asynccnt(2);
#  else
    asm volatile("s_wait_asynccnt 0x2" ::: "memory");
#  endif
#endif
}

// wave-local LDS ordering point: no ISA emitted, hardware keeps a wave's
// LDS ops in order; this only pins the compiler's memory/scheduling order.
__device__ __forceinline__ void wave_lds_fence() {
    __builtin_amdgcn_wave_barrier();
    __builtin_amdgcn_fence(__ATOMIC_SEQ_CST, "wavefront");
    __builtin_amdgcn_wave_barrier();
}

#define W_IMG 512
#define GPW   8        // 4-block groups per wave (32 blocks, 8KB traffic)

__global__ __launch_bounds__(256) void dct8_roundtrip_wmma(
    const float* __restrict__ in, float* __restrict__ out, int nGroups)
{
    // two linear 256-float tiles per wave (double buffer); a group is an
    // 8x32 contiguous-row region: 4 width-consecutive blocks share rows.
    __shared__ float tile[8][2][256];
    const int tid  = threadIdx.x;
    const int wave = tid >> 5;
    const int lane = tid & 31;
    const int h    = lane >> 4;    // lane half (K pair / M+8 rows)
    const int n    = lane & 15;    // N column for B/C/D, M row for A

    // ---- constant B fragments, built once per wave -------------------------
    // Forward: Bd[j][n] = same-half ? 2*cos(pi*(2j+1)*n/16) : 0
    // Inverse: Bi[j][n] = same-half ? 2*s_j*cos(pi*(2n+1)*j/16) : 0,
    //          s_0=1/32, s_j=1/16  (exact inverse of the x2 DCT-II)
    const float PI16 = 0.19634954084936207f;
    float bF[4][2], bI[4][2];
#pragma unroll
    for (int t = 0; t < 4; ++t) {
#pragma unroll
        for (int u = 0; u < 2; ++u) {
            const int  j    = 4 * t + 2 * h + u;
            const bool same = ((j < 8) == (n < 8));
            const int  jj   = j & 7, nn = n & 7;
            bF[t][u] = same ? 2.0f * __cosf(PI16 * (float)((2 * jj + 1) * nn)) : 0.0f;
            const float sj = (jj == 0) ? 0.03125f : 0.0625f;
            bI[t][u] = same ? 2.0f * sj * __cosf(PI16 * (float)((2 * nn + 1) * jj)) : 0.0f;
        }
    }

    // A-fragment addresses in the linear tile: element (m,c) of the 16x16
    // logical tile lives at L[(m&7)*32 + (2*(c>=8)+(m>=8))*8 + (c&7)]
    int fragIdx[4];
#pragma unroll
    for (int t = 0; t < 4; ++t) {
        const int c0 = 4 * t + 2 * h;
        const int gg = ((c0 >= 8) ? 2 : 0) + ((n >= 8) ? 1 : 0);
        fragIdx[t] = (n & 7) * 32 + gg * 8 + (c0 & 7);
    }
    const int ggY = ((n >= 8) ? 2 : 0) + h;   // local block for D-layout rows
    const int nn7 = n & 7;

    // issue one group's stage-in (2 async b128 per lane) into a tile buffer
    auto stage_in = [&](int gw2, float* Tdst) {
        const int gb  = gw2 << 2;
        const int img = gb >> 12, ib = gb & 4095;
        const float* src = in + (size_t)img * (W_IMG * W_IMG)
                              + (size_t)((ib >> 6) << 3) * W_IMG
                              + ((ib & 63) << 3);
#pragma unroll
        for (int q = 0; q < 2; ++q) {
            const int f  = lane + (q << 5);     // float4 index 0..63
            const int r  = f >> 3;              // row 0..7
            const int ch = f & 7;               // 16B chunk in row
            const float* g = src + (size_t)r * W_IMG + (ch << 2);
#if defined(HAS_ASYNC_LDS)
            __builtin_amdgcn_global_load_async_to_lds_b128(
                (g_v4i*)g, (l_v4i*)&Tdst[f << 2], 0, 0);
#else
            *reinterpret_cast<v4f*>(&Tdst[f << 2]) =
                *reinterpret_cast<const v4f*>(g);
#endif
        }
    };

    const int gBase = (blockIdx.x * 8 + wave) * GPW;

    // prime the pipeline: group 0 into buffer 0
    if (gBase < nGroups) stage_in(gBase, tile[wave][0]);

    for (int it = 0; it < GPW; ++it) {
        const int  gw     = gBase + it;
        const bool active = (gw < nGroups);               // wave-uniform
        float* T  = tile[wave][it & 1];
        float* Tn = tile[wave][(it + 1) & 1];

        // issue next group's loads into the other buffer, then drain current
        const bool nextIssued = (it + 1 < GPW) && (gw + 1 < nGroups);
        if (nextIssued) stage_in(gw + 1, Tn);
        if (nextIssued) wait_async_keep2(); else wait_async_all();
        wave_lds_fence();

        // prefetch at distance 2 (8 lanes, one aligned 128B row each)
        const int gp = gw + 2;
        if (gp < nGroups && it + 2 < GPW && lane < 8) {
            const int gb2  = gp << 2;
            const int img2 = gb2 >> 12, ib2 = gb2 & 4095;
            const float* p = in + (size_t)img2 * (W_IMG * W_IMG)
                           + (size_t)(((ib2 >> 6) << 3) + lane) * W_IMG
                           + ((ib2 & 63) << 3);
            __builtin_prefetch(p, 0, 3);
        }

        // ---- stage 1: Y = A x blockdiag(C^T, C^T) --------------------------
        v8f acc = {};
#pragma unroll
        for (int t = 0; t < 4; ++t) {
            v2f a; a.x = T[fragIdx[t]]; a.y = T[fragIdx[t] + 1];
            v2f b; b.x = bF[t][0];      b.y = bF[t][1];
            acc = __builtin_amdgcn_wmma_f32_16x16x4_f32(
                false, a, false, b, (short)0, acc, false, false);
        }
        wave_lds_fence();
#pragma unroll
        for (int rr = 0; rr < 8; ++rr)        // D layout -> same linear map
            T[rr * 32 + ggY * 8 + nn7] = acc[rr];
        wave_lds_fence();

        // ---- stage 2: Z = Y x blockdiag(Cinv^T, Cinv^T) --------------------
        v8f accZ = {};
#pragma unroll
        for (int t = 0; t < 4; ++t) {
            v2f a; a.x = T[fragIdx[t]]; a.y = T[fragIdx[t] + 1];
            v2f b; b.x = bI[t][0];      b.y = bI[t][1];
            accZ = __builtin_amdgcn_wmma_f32_16x16x4_f32(
                false, a, false, b, (short)0, accZ, false, false);
        }
        wave_lds_fence();
#pragma unroll
        for (int rr = 0; rr < 8; ++rr)        // D layout -> block-major linear
            T[ggY * 64 + rr * 8 + nn7] = accZ[rr];
        wave_lds_fence();

        // ---- stage out: 1KB contiguous block-major region, NT b128 stores --
        if (active) {
            float* dst = out + (size_t)gw * 256;
#pragma unroll
            for (int q = 0; q < 2; ++q) {
                const int f = lane + (q << 5);
                const v4f v = *reinterpret_cast<const v4f*>(&T[f << 2]);
                __builtin_nontemporal_store(
                    v, reinterpret_cast<v4f*>(dst + (f << 2)));
            }
        }
        wave_lds_fence();   // buffer reused two iterations later
    }
}

extern "C" void kernel_launch(void* const* d_in, const int* in_sizes, int n_in,
                              void* d_out, int out_size, void* d_ws, size_t ws_size,
                              hipStream_t stream) {
    (void)n_in; (void)out_size; (void)d_ws; (void)ws_size;
    const float* x  = (const float*)d_in[0];
    float* out      = (float*)d_out;
    const int nGroups = in_sizes[0] >> 8;            // 256 floats per group
    const int ctas    = (nGroups + 8 * GPW - 1) / (8 * GPW);
    dct8_roundtrip_wmma<<<ctas, 256, 0, stream>>>(x, out, nGroups);
}